// CapsuleLayer_4063039062212
// MI455X (gfx1250) — compile-verified
//
#include <hip/hip_runtime.h>
#include <hip/hip_bf16.h>
#include <math.h>

// Problem constants (from reference)
#define B_DIM 8
#define I_DIM 32
#define C_DIM 8
#define O_DIM 16
#define H_POSE 16
#define PIX 256           // 16x16
#define M_DIM 256         // O_DIM * H_POSE
#define NUM_ROUTING 3

typedef __attribute__((ext_vector_type(2))) float v2f;
typedef __attribute__((ext_vector_type(8))) float v8f;

// ---------------------------------------------------------------------------
// Kernel 1: votes[b,p,i,m] = sum_c x[b,i,c,p] * w[i,c,m]
// One wave per (i, b, 16-pixel block). M=16 pixels, K=8 (2x WMMA K=4), N=256.
// Uses V_WMMA_F32_16X16X4_F32 (full f32 precision, matches reference).
// ---------------------------------------------------------------------------
__global__ __launch_bounds__(256) void capsule_votes_wmma(
    const float* __restrict__ x,      // [B,I,C,256]
    const float* __restrict__ w,      // [I,C,256]
    float* __restrict__ votes)        // [B,256,I,256]
{
    const int lane  = threadIdx.x & 31;
    const int wave  = blockIdx.x * (blockDim.x >> 5) + (threadIdx.x >> 5);
    const int i     = wave & (I_DIM - 1);       // 0..31
    const int pbg   = wave >> 5;                // 0..127
    const int b     = pbg >> 4;                 // 0..7
    const int pblk  = pbg & 15;                 // 0..15 (16-pixel block)
    const int lo    = lane & 15;
    const int hi    = lane >> 4;                // 0 or 1

    // --- A fragment: 16x4 f32, rows = pixels, K = channel c ---
    // VGPR0: K=0 (lanes 0-15) / K=2 (lanes 16-31); VGPR1: K=1 / K=3
    const int p = pblk * 16 + lo;               // pixel row M = lo
    const float* xb = x + ((size_t)(b * I_DIM + i) * C_DIM) * PIX + p;
    v2f a0, a1;
    a0.x = xb[(2 * hi + 0) * PIX];
    a0.y = xb[(2 * hi + 1) * PIX];
    a1.x = xb[(2 * hi + 4) * PIX];
    a1.y = xb[(2 * hi + 5) * PIX];

    const float* wi = w + (size_t)i * C_DIM * M_DIM;
    // votes base for (b, pixel-block start, i)
    float* vout = votes + ((size_t)(b * PIX + pblk * 16) * I_DIM + i) * M_DIM;

    for (int nt = 0; nt < 16; ++nt) {
        const int n = nt * 16 + lo;             // output column N = lo
        // --- B fragment: 4x16 f32, rows = channel c, cols = m ---
        v2f b0, b1;
        b0.x = wi[(2 * hi + 0) * M_DIM + n];
        b0.y = wi[(2 * hi + 1) * M_DIM + n];
        b1.x = wi[(2 * hi + 4) * M_DIM + n];
        b1.y = wi[(2 * hi + 5) * M_DIM + n];

        v8f acc = {};
        acc = __builtin_amdgcn_wmma_f32_16x16x4_f32(
            false, a0, false, b0, (short)0, acc, false, false);
        acc = __builtin_amdgcn_wmma_f32_16x16x4_f32(
            false, a1, false, b1, (short)0, acc, false, false);

        // D layout: lane group 'hi' holds rows r+8*hi, col = lo
#pragma unroll
        for (int r = 0; r < 8; ++r) {
            const int row = r + 8 * hi;         // pixel within block
            vout[(size_t)row * (I_DIM * M_DIM) + nt * 16 + lo] = acc[r];
        }
    }
}

// ---------------------------------------------------------------------------
// LDS byte-offset helper (for async-to-LDS vdst operand)
// ---------------------------------------------------------------------------
__device__ __forceinline__ unsigned lds_byte_off(const void* p) {
    return (unsigned)(unsigned long long)
        (__attribute__((address_space(3))) const char*)p;
}

// ---------------------------------------------------------------------------
// Kernel 2: full 3-iteration dynamic routing. One workgroup per (b, pixel).
// Votes for this pixel (32KB) pulled to LDS with global_load_async_to_lds_b128.
// Shared layout (floats): [0,8192) votes[i][o][h'] | [8192,8704) logits[i][o]
//                         [8704,9216) route[i][o]  | [9216,9472) act[o][h']
// ---------------------------------------------------------------------------
#define V_OFF 0
#define L_OFF 8192
#define R_OFF 8704
#define A_OFF 9216

__global__ __launch_bounds__(256) void capsule_routing(
    const float* __restrict__ votes,   // [B,256,I,256]
    const float* __restrict__ bias,    // [O,h] = 256
    float* __restrict__ out)           // [B,O,h,256]
{
    __shared__ float smem[9472];

    const int t  = threadIdx.x;        // 0..255
    const int bp = blockIdx.x;         // b*256 + p
    const int b  = bp >> 8;
    const int p  = bp & 255;

    // ---- async copy votes[b,p,:,:] (8192 floats) into LDS ----
    const float* src = votes + (size_t)bp * (I_DIM * M_DIM);
    const unsigned lds0 = lds_byte_off(&smem[V_OFF]);
#pragma unroll
    for (int k = 0; k < 8; ++k) {
        const unsigned byte_off = (unsigned)(t + k * 256) * 16u;  // B128 chunks
        const unsigned ldsa = lds0 + byte_off;
        asm volatile("global_load_async_to_lds_b128 %0, %1, %2"
                     :
                     : "v"(ldsa), "v"(byte_off), "s"(src)
                     : "memory");
    }

    // zero logits while the DMA is in flight
    smem[L_OFF + t]       = 0.0f;
    smem[L_OFF + 256 + t] = 0.0f;

    asm volatile("s_wait_asynccnt 0" ::: "memory");
    __syncthreads();

    const int o  = t >> 4;             // 0..15 output capsule
    const int hp = t & 15;             // 0..15 pose element
    const float bia = bias[t];         // bias[o*16+h']

    float act = 0.0f;
    for (int it = 0; it < NUM_ROUTING; ++it) {
        // ---- softmax over o (per i): threads 0..31, one capsule-input each ----
        if (t < I_DIM) {
            float mx = smem[L_OFF + t * O_DIM];
            for (int oo = 1; oo < O_DIM; ++oo)
                mx = fmaxf(mx, smem[L_OFF + t * O_DIM + oo]);
            float sum = 0.0f;
            float e[O_DIM];
            for (int oo = 0; oo < O_DIM; ++oo) {
                e[oo] = __expf(smem[L_OFF + t * O_DIM + oo] - mx);
                sum += e[oo];
            }
            const float inv = 1.0f / sum;
            for (int oo = 0; oo < O_DIM; ++oo)
                smem[R_OFF + t * O_DIM + oo] = e[oo] * inv;
        }
        __syncthreads();

        // ---- preactivate[o,h'] = sum_i route[i,o] * votes[i,o,h'] + bias ----
        float pre = bia;
#pragma unroll 8
        for (int i = 0; i < I_DIM; ++i)
            pre += smem[R_OFF + i * O_DIM + o] * smem[V_OFF + i * M_DIM + t];

        // ---- squash over h' (16 contiguous lanes of a wave32) ----
        float nsq = pre * pre;
        nsq += __shfl_xor(nsq, 1, 16);
        nsq += __shfl_xor(nsq, 2, 16);
        nsq += __shfl_xor(nsq, 4, 16);
        nsq += __shfl_xor(nsq, 8, 16);
        // s/|s| * |s|^2/(1+|s|^2) == s * |s| / (1+|s|^2)
        act = pre * __fsqrt_rn(nsq) / (1.0f + nsq);

        if (it == NUM_ROUTING - 1) break;

        smem[A_OFF + t] = act;
        __syncthreads();

        // ---- logits[i,o] += sum_h votes[i,o,h] * act[o,h] ----
#pragma unroll
        for (int io = t; io < I_DIM * O_DIM; io += 256) {
            const int oo = io & (O_DIM - 1);
            float d = 0.0f;
#pragma unroll
            for (int h = 0; h < H_POSE; ++h)
                d += smem[V_OFF + io * H_POSE + h] * smem[A_OFF + oo * H_POSE + h];
            smem[L_OFF + io] += d;
        }
        __syncthreads();
    }

    // out[b, o, h', p]
    out[(size_t)(b * M_DIM + t) * PIX + p] = act;
}

// ---------------------------------------------------------------------------
extern "C" void kernel_launch(void* const* d_in, const int* in_sizes, int n_in,
                              void* d_out, int out_size, void* d_ws, size_t ws_size,
                              hipStream_t stream) {
    const float* x    = (const float*)d_in[0];   // [8,32,8,16,16]
    const float* w    = (const float*)d_in[1];   // [32,8,256]
    const float* bias = (const float*)d_in[2];   // [16,16,1,1]
    float* out   = (float*)d_out;                // [8,16,16,16,16]
    float* votes = (float*)d_ws;                 // needs 64 MB: [8,256,32,256]

    // 4096 waves (I * B * 16 pixel-blocks), 8 waves per block
    capsule_votes_wmma<<<512, 256, 0, stream>>>(x, w, votes);
    // one workgroup per (b, pixel)
    capsule_routing<<<B_DIM * PIX, 256, 0, stream>>>(votes, bias, out);
}